// SwitchTransformersTop1Router_10831907520600
// MI455X (gfx1250) — compile-verified
//
#include <hip/hip_runtime.h>
#include <hip/hip_bf16.h>

typedef float v2f __attribute__((ext_vector_type(2)));
typedef float v8f __attribute__((ext_vector_type(8)));

#define NE   8
#define HID  1024
#define SEQ  4096
#define NB   8
#define TOKENS (NB * SEQ)      // 32768
#define CAP  640
#define WROW 1028              // padded LDS row stride (floats): kills bank conflicts

// ---------------------------------------------------------------------------
// Phase 1: router logits via V_WMMA_F32_16X16X4_F32, softmax max-prob + argmax
// Grid: 256 blocks x 256 threads (8 waves/block, 16 tokens/wave, 128 tokens/block)
// ---------------------------------------------------------------------------
__launch_bounds__(256)
__global__ void router_phase1_kernel(const float* __restrict__ H,
                                     const float* __restrict__ W,
                                     float* __restrict__ out,
                                     int* __restrict__ idx_ws)
{
    // W staged transposed+padded: ldsW[e][k], e in 0..8 (row 8 = zeros for pad lanes)
    __shared__ float ldsW[9 * WROW];
    __shared__ float ldsLog[8 * 16 * NE];   // per-wave 16 tokens x 8 logits

    const int tid = threadIdx.x;

    // Stage W (1024x8 row-major) into LDS transposed; row 8 zero-filled.
    for (int i = tid; i < 9 * HID; i += 256) {
        int e = i >> 10;          // 0..8
        int k = i & (HID - 1);
        ldsW[e * WROW + k] = (e < NE) ? W[k * NE + e] : 0.0f;
    }
    __syncthreads();

    const int wave  = tid >> 5;
    const int lane  = tid & 31;
    const int half  = lane >> 4;        // 0: K+0/K+1, 1: K+2/K+3
    const int n     = lane & 15;        // A: row M; B/C: column N
    const int token0 = blockIdx.x * 128 + wave * 16;

    const float* hrow = H + (size_t)(token0 + n) * HID + half * 2;
    const float* wrow = ldsW + (size_t)((n < NE) ? n : NE) * WROW + half * 2;

    v8f c = {};
#pragma unroll 8
    for (int k = 0; k < HID; k += 4) {
        v2f a = *(const v2f*)(hrow + k);          // A 16x4 f32 fragment (2 VGPRs)
        v2f b;                                     // B 4x16 f32 fragment (2 VGPRs)
        b.x = wrow[k];
        b.y = wrow[k + 1];
        // D = A*B + C : v_wmma_f32_16x16x4_f32
        c = __builtin_amdgcn_wmma_f32_16x16x4_f32(
                /*neg_a=*/false, a, /*neg_b=*/false, b,
                /*c_mod=*/(short)0, c, /*reuse_a=*/false, /*reuse_b=*/false);
    }

    // Scatter C (VGPR j -> M=j+8*half, N=lane&15) so one lane owns one token.
    float* wls = ldsLog + wave * (16 * NE);
    if (n < NE) {
#pragma unroll
        for (int j = 0; j < 8; ++j) {
            int m = j + half * 8;
            wls[m * NE + n] = c[j];
        }
    }
    __syncthreads();

    if (lane < 16) {
        const int token = token0 + lane;
        const float* lg = ldsLog + wave * (16 * NE) + lane * NE;
        // first-max argmax (matches jnp.argmax tie-break), softmax max-prob
        float mx = lg[0];
        int   am = 0;
#pragma unroll
        for (int e = 1; e < NE; ++e) {
            float v = lg[e];
            if (v > mx) { mx = v; am = e; }
        }
        float s = 0.0f;
#pragma unroll
        for (int e = 0; e < NE; ++e) s += expf(lg[e] - mx);
        float p = 1.0f / s;

        out[token] = p;                                 // output 0: max_probs
        out[TOKENS + TOKENS * NE + token] = p;          // output 2: max_probs (again)
        idx_ws[token] = am;
    }
}

// ---------------------------------------------------------------------------
// Phase 2: capacity mask via wave32 ballot scan (cumsum of one-hot per batch)
// Grid: 8 blocks (one per batch) x 32 threads (one wave)
// ---------------------------------------------------------------------------
__launch_bounds__(32)
__global__ void router_phase2_kernel(const int* __restrict__ idx_ws,
                                     float* __restrict__ out_mask)
{
    const int b    = blockIdx.x;
    const int lane = threadIdx.x;
    const unsigned lt = (1u << lane) - 1u;

    const int* idx = idx_ws + b * SEQ;
    float* om = out_mask + (size_t)b * SEQ * NE;

    unsigned counts[NE];
#pragma unroll
    for (int e = 0; e < NE; ++e) counts[e] = 0u;

    for (int s0 = 0; s0 < SEQ; s0 += 32) {
        int my = idx[s0 + lane];
        int keep = 0;
#pragma unroll
        for (int e = 0; e < NE; ++e) {
            unsigned bal = (unsigned)__ballot(my == e);   // wave32: low 32 bits
            if (my == e) {
                unsigned prio = counts[e] + __popc(bal & lt) + 1u;  // 1-based rank
                keep = (prio <= CAP) ? 1 : 0;
            }
            counts[e] += __popc(bal);
        }
        float* row = om + (size_t)(s0 + lane) * NE;
#pragma unroll
        for (int e = 0; e < NE; ++e)
            row[e] = (e == my && keep) ? 1.0f : 0.0f;
    }
}

extern "C" void kernel_launch(void* const* d_in, const int* in_sizes, int n_in,
                              void* d_out, int out_size, void* d_ws, size_t ws_size,
                              hipStream_t stream) {
    const float* H = (const float*)d_in[0];   // (8, 4096, 1024) f32
    const float* W = (const float*)d_in[1];   // (1024, 8) f32
    float* out = (float*)d_out;               // [32768 | 262144 | 32768] floats
    int* idx_ws = (int*)d_ws;                 // 32768 ints of scratch

    router_phase1_kernel<<<256, 256, 0, stream>>>(H, W, out, idx_ws);
    router_phase2_kernel<<<NB, 32, 0, stream>>>(idx_ws, out + TOKENS);
}